// Siamese_GeoChebyConv_70849780514832
// MI455X (gfx1250) — compile-verified
//
#include <hip/hip_runtime.h>

typedef __attribute__((ext_vector_type(2))) float v2f;
typedef __attribute__((ext_vector_type(4))) float v4f;
typedef __attribute__((ext_vector_type(8))) float v8f;

#define N_NODES 268
#define N_PAD   272
#define E_EDGES 8192
#define F_IN    268
#define NHID    64
#define NCLS    32
#define BLK     256
#define NWAVE   8

// LDS strides (floats). 36 and 68: rows 16B-aligned, bank-conflict-free for
// the column-indexed WMMA A-fragment pattern.
#define SB  68    // big buffers [272 x 64]
#define SC  36    // chunk / h2 buffers [272 x <=32]
#define SZ  104   // z buffer [32 x 100(+pad)]
#define SZ2 64    // z2 buffer [32 x 60(+pad)]

// LDS layout (float offsets)
#define OFF_NW   0
#define OFF_EP   8192
#define OFF_DEG  16384
#define OFF_B0   16704
#define OFF_B1   (OFF_B0 + N_PAD * SB)
#define OFF_B2   (OFF_B1 + N_PAD * SB)
#define LDS_FLOATS (OFF_B2 + N_PAD * SB)   // 72192 floats
#define LDS_BYTES  (LDS_FLOATS * 4)        // 288768 bytes < 320KB

// ---- CDNA5 f32 WMMA: D[16x16] = A[16x4] * B[4x16] + C ----
__device__ __forceinline__ v8f wmma4(v2f a, v2f b, v8f c) {
  return __builtin_amdgcn_wmma_f32_16x16x4_f32(false, a, false, b, (short)0, c,
                                               false, false);
}

// CDNA5 async global->LDS copy (16B per lane), tracked by ASYNCcnt.
__device__ __forceinline__ void async_b128_to_lds(unsigned lds_off,
                                                  const void* gptr) {
  asm volatile("global_load_async_to_lds_b128 %0, %1, off"
               :: "v"(lds_off), "v"(gptr) : "memory");
}
__device__ __forceinline__ void wait_async0() {
  asm volatile("s_wait_asynccnt 0x0" ::: "memory");
}

// A-fragment (16x4, row-major source, stride in floats):
// lanes 0-15: M=lane, K = k0,k0+1 ; lanes 16-31: M=lane-16, K = k0+2,k0+3
__device__ __forceinline__ v2f afrag(const float* m, int stride, int mbase,
                                     int k0, int lane) {
  int r = mbase + (lane & 15);
  int k = k0 + ((lane >> 4) << 1);
  const float* p = m + r * stride + k;
  v2f a; a.x = p[0]; a.y = p[1];
  return a;
}

// A-fragment from a transposed view: logical A[M][K] = src[K][M]
__device__ __forceinline__ v2f afragT(const float* m, int stride, int mbase,
                                      int k0, int lane) {
  int c = mbase + (lane & 15);
  int k = k0 + ((lane >> 4) << 1);
  v2f a; a.x = m[k * stride + c]; a.y = m[(k + 1) * stride + c];
  return a;
}

// B-fragment (4x16) from global row-major W[K][ncols], column-guarded
__device__ __forceinline__ v2f bfrag(const float* W, int ncols, int kbase,
                                     int nbase, int lane, int nmax) {
  int n = nbase + (lane & 15);
  int k = kbase + ((lane >> 4) << 1);
  bool ok = (n < nmax);
  v2f b;
  b.x = ok ? W[k * ncols + n] : 0.f;
  b.y = ok ? W[(k + 1) * ncols + n] : 0.f;
  return b;
}

// C/D layout store: VGPR j, lanes0-15 -> (M=j, N=lane); lanes16-31 -> (M=8+j)
__device__ __forceinline__ void store_tile(float* dst, int stride, int mbase,
                                           int nbase, int lane, v8f acc,
                                           const float* bias, int nmax,
                                           int mmax, bool do_relu) {
  int n = nbase + (lane & 15);
  int mo = (lane >> 4) << 3;
  float bv = (n < nmax) ? bias[n] : 0.f;
#pragma unroll
  for (int j = 0; j < 8; ++j) {
    int m = mbase + mo + j;
    float v = acc[j] + bv;
    if (do_relu) v = fmaxf(v, 0.f);
    if (m < mmax && n < nmax) dst[m * stride + n] = v;
  }
}

// out[dst,:] += nw[e] * in[src,:]  via LDS float atomics (ds_add_f32)
// nf must be a multiple of 4; row bases are 16B aligned.
__device__ __forceinline__ void lx_scatter(float* outb, const float* inb,
                                           int stride, int nf,
                                           const unsigned* ep, const float* nw,
                                           int tid) {
  for (int e = tid; e < E_EDGES; e += BLK) {
    unsigned p = ep[e];
    float c = nw[e];
    int s = (int)(p & 0xFFFFu), d = (int)(p >> 16);
    const float* srow = inb + s * stride;
    float* drow = outb + d * stride;
    for (int f = 0; f < nf; f += 4) {
      v4f sv = *(const v4f*)(srow + f);
      atomicAdd(&drow[f + 0], c * sv.x);
      atomicAdd(&drow[f + 1], c * sv.y);
      atomicAdd(&drow[f + 2], c * sv.z);
      atomicAdd(&drow[f + 3], c * sv.w);
    }
  }
}

extern "C" __global__ void __launch_bounds__(BLK, 1)
siamese_cheb_kernel(const float* __restrict__ x1, const int* __restrict__ ei1,
                    const float* __restrict__ ea1, const float* __restrict__ x2,
                    const int* __restrict__ ei2, const float* __restrict__ ea2,
                    const float* __restrict__ W1, const float* __restrict__ b1,
                    const float* __restrict__ W4, const float* __restrict__ b4,
                    const float* __restrict__ Wc1, const float* __restrict__ bc1,
                    const float* __restrict__ Wc2, const float* __restrict__ bc2,
                    const float* __restrict__ Wc3, const float* __restrict__ bc3,
                    float* __restrict__ out) {
  extern __shared__ float sm[];
  const int tid = threadIdx.x;
  const int lane = tid & 31;
  // wave id is wave-uniform: pin it to an SGPR so tile guards are scalar
  // branches and EXEC stays all-ones around WMMA.
  const int wid = __builtin_amdgcn_readfirstlane(tid >> 5);
  const int blk = blockIdx.x;     // 0..127
  const int g = blk & 63;
  const int side = blk >> 6;

  const float* xg = (side ? x2 : x1) + (size_t)g * N_NODES * F_IN;
  const int* srcg = (side ? ei2 : ei1) + (size_t)g * 2 * E_EDGES;
  const int* dstg = srcg + E_EDGES;
  const float* eag = (side ? ea2 : ea1) + (size_t)g * E_EDGES;

  float* nw = sm + OFF_NW;
  unsigned* ep = (unsigned*)(sm + OFF_EP);
  float* deg = sm + OFF_DEG;
  float* B0 = sm + OFF_B0;
  float* B1 = sm + OFF_B1;
  float* B2 = sm + OFF_B2;

  // LDS byte offset of the dynamic shared base (addrspace(3) view).
  typedef __attribute__((address_space(3))) char lds_char;
  const unsigned lds_base = (unsigned)(size_t)(lds_char*)(void*)sm;

  const v4f z4 = {0.f, 0.f, 0.f, 0.f};

  // ================= edge prep: deg, dinv, nw =================
  for (int n = tid; n < 320; n += BLK) deg[n] = 0.f;
  __syncthreads();
  for (int e = tid; e < E_EDGES; e += BLK) {
    int s = srcg[e], d = dstg[e];
    float w = eag[e];
    if (s == d) w = 0.f;                       // remove_self_loops
    ep[e] = (unsigned)s | ((unsigned)d << 16); // N=268 fits 16 bits
    nw[e] = w;
    atomicAdd(&deg[s], w);
  }
  __syncthreads();
  for (int n = tid; n < N_NODES; n += BLK) {
    float dv = deg[n];
    deg[n] = (dv > 0.f) ? rsqrtf(dv) : 0.f;
  }
  __syncthreads();
  for (int e = tid; e < E_EDGES; e += BLK) {
    unsigned p = ep[e];
    int s = (int)(p & 0xFFFFu), d = (int)(p >> 16);
    nw[e] = -deg[s] * nw[e] * deg[d];
  }

  // ================= gc1: h1 = relu(cheb(x, W1) + b1), feature-chunked ======
  {
    v8f acc[9];
    const v8f z8 = {0, 0, 0, 0, 0, 0, 0, 0};
#pragma unroll
    for (int i = 0; i < 9; ++i) acc[i] = z8;

    float* xb = B0;    // [272 x SC] chunk of x
    float* t1b = B1;   // T1 = L_hat x
    float* t2b = B2;   // T2 = 2 L_hat T1 - x

    for (int c0 = 0; c0 < F_IN; c0 += 32) {
      const int CS = (F_IN - c0 < 32) ? (F_IN - c0) : 32;   // 32,...,32,12
      const int n4 = CS >> 2;
      __syncthreads();
      // async copy of the x chunk: 16B packets, ASYNCcnt-tracked
      for (int r = tid; r < N_NODES; r += BLK) {
        unsigned loff = lds_base + (unsigned)((OFF_B0 + r * SC) * 4);
        const float* gp = xg + (size_t)r * F_IN + c0;
        for (int c4 = 0; c4 < n4; ++c4)
          async_b128_to_lds(loff + 16u * c4, gp + 4 * c4);
      }
      for (int r = N_NODES + tid; r < N_PAD; r += BLK) {   // zero pad rows
        v4f* row = (v4f*)(xb + r * SC);
        for (int q = 0; q < n4; ++q) row[q] = z4;
      }
      {
        v4f* t14 = (v4f*)t1b;
        for (int i = tid; i < (N_PAD * SC) >> 2; i += BLK) t14[i] = z4;
      }
      wait_async0();
      __syncthreads();
      lx_scatter(t1b, xb, SC, CS, ep, nw, tid);
      __syncthreads();
      {
        v4f* t24 = (v4f*)t2b;
        for (int i = tid; i < (N_PAD * SC) >> 2; i += BLK) t24[i] = z4;
      }
      __syncthreads();
      lx_scatter(t2b, t1b, SC, CS, ep, nw, tid);
      __syncthreads();
      for (int r = tid; r < N_NODES; r += BLK) {
        v4f* t2r = (v4f*)(t2b + r * SC);
        const v4f* xr = (const v4f*)(xb + r * SC);
        for (int q = 0; q < n4; ++q) t2r[q] = 2.f * t2r[q] - xr[q];
      }
      __syncthreads();
      // accumulate [272 x CS] @ [CS x 64] for 3 Chebyshev terms.
      // Static accumulator index i keeps acc[] in VGPRs (no scratch).
#pragma unroll
      for (int i = 0; i < 9; ++i) {
        const int t = wid + NWAVE * i;          // 17 M-tiles x 4 N-tiles = 68
        if (t < 68) {
          const int mb = (t >> 2) << 4;
          const int nb = (t & 3) << 4;
          v8f a_ = acc[i];
          for (int k = 0; k < CS; k += 4) {
            v2f ax = afrag(xb, SC, mb, k, lane);
            v2f a1 = afrag(t1b, SC, mb, k, lane);
            v2f a2 = afrag(t2b, SC, mb, k, lane);
            const int kg = c0 + k;
            v2f w0 = bfrag(W1, NHID, kg, nb, lane, NHID);
            v2f w1 = bfrag(W1 + F_IN * NHID, NHID, kg, nb, lane, NHID);
            v2f w2 = bfrag(W1 + 2 * F_IN * NHID, NHID, kg, nb, lane, NHID);
            a_ = wmma4(ax, w0, a_);
            a_ = wmma4(a1, w1, a_);
            a_ = wmma4(a2, w2, a_);
          }
          acc[i] = a_;
        }
      }
    }
    __syncthreads();
    // h1 -> B0 [268 x 64], stride SB (xb is dead)
#pragma unroll
    for (int i = 0; i < 9; ++i) {
      const int t = wid + NWAVE * i;
      if (t < 68)
        store_tile(B0, SB, (t >> 2) << 4, (t & 3) << 4, lane, acc[i], b1,
                   NHID, N_NODES, true);
    }
  }
  __syncthreads();

  // ================= gc4: h2 = relu(cheb(h1, W4) + b4) =====================
  {
    {
      v4f* b14 = (v4f*)B1;
      for (int i = tid; i < (N_PAD * SB) >> 2; i += BLK) b14[i] = z4;
    }
    __syncthreads();
    lx_scatter(B1, B0, SB, NHID, ep, nw, tid);
    __syncthreads();
    {
      v4f* b24 = (v4f*)B2;
      for (int i = tid; i < (N_PAD * SB) >> 2; i += BLK) b24[i] = z4;
    }
    __syncthreads();
    lx_scatter(B2, B1, SB, NHID, ep, nw, tid);
    __syncthreads();
    for (int r = tid; r < N_NODES; r += BLK) {
      v4f* t2r = (v4f*)(B2 + r * SB);
      const v4f* hr = (const v4f*)(B0 + r * SB);
      for (int q = 0; q < (NHID >> 2); ++q) t2r[q] = 2.f * t2r[q] - hr[q];
    }
    __syncthreads();

    v8f acc2[5];
    const v8f z8 = {0, 0, 0, 0, 0, 0, 0, 0};
#pragma unroll
    for (int i = 0; i < 5; ++i) acc2[i] = z8;
#pragma unroll
    for (int i = 0; i < 5; ++i) {
      const int t = wid + NWAVE * i;            // 17 x 2 tiles = 34
      if (t < 34) {
        const int mb = (t >> 1) << 4;
        const int nb = (t & 1) << 4;
        v8f a_ = acc2[i];
        for (int k = 0; k < NHID; k += 4) {
          v2f ah = afrag(B0, SB, mb, k, lane);
          v2f a1 = afrag(B1, SB, mb, k, lane);
          v2f a2 = afrag(B2, SB, mb, k, lane);
          v2f w0 = bfrag(W4, NCLS, k, nb, lane, NCLS);
          v2f w1 = bfrag(W4 + NHID * NCLS, NCLS, k, nb, lane, NCLS);
          v2f w2 = bfrag(W4 + 2 * NHID * NCLS, NCLS, k, nb, lane, NCLS);
          a_ = wmma4(ah, w0, a_);
          a_ = wmma4(a1, w1, a_);
          a_ = wmma4(a2, w2, a_);
        }
        acc2[i] = a_;
      }
    }
    __syncthreads();
    // h2 -> B1 [268 x 32], stride SC (t1 dead)
#pragma unroll
    for (int i = 0; i < 5; ++i) {
      const int t = wid + NWAVE * i;
      if (t < 34)
        store_tile(B1, SC, (t >> 1) << 4, (t & 1) << 4, lane, acc2[i], b4,
                   NCLS, N_NODES, true);
    }
  }
  __syncthreads();

  // ============ classifier L1: z = relu(h2^T @ Wc1 + bc1)  [32 x 100] ======
  {
    v8f accc[2];
    const v8f z8 = {0, 0, 0, 0, 0, 0, 0, 0};
    accc[0] = z8; accc[1] = z8;
#pragma unroll
    for (int i = 0; i < 2; ++i) {
      const int t = wid + NWAVE * i;            // 2 x 7 tiles = 14 (N pad 112)
      if (t < 14) {
        const int mb = (t / 7) << 4;
        const int nb = (t % 7) << 4;
        v8f a_ = accc[i];
        for (int k = 0; k < N_NODES; k += 4) {  // K = 268
          v2f a = afragT(B1, SC, mb, k, lane);
          v2f b = bfrag(Wc1, 100, k, nb, lane, 100);
          a_ = wmma4(a, b, a_);
        }
        accc[i] = a_;
      }
    }
    __syncthreads();
#pragma unroll
    for (int i = 0; i < 2; ++i) {               // z -> B2 (t2 dead)
      const int t = wid + NWAVE * i;
      if (t < 14)
        store_tile(B2, SZ, (t / 7) << 4, (t % 7) << 4, lane, accc[i], bc1,
                   100, NCLS, true);
    }
  }
  __syncthreads();

  // ============ classifier L2: z2 = relu(z @ Wc2 + bc2)  [32 x 60] =========
  {
    const int t = wid;                          // exactly 8 tiles (2x4)
    const int mb = (t >> 2) << 4;
    const int nb = (t & 3) << 4;
    v8f a_ = {0, 0, 0, 0, 0, 0, 0, 0};
    for (int k = 0; k < 100; k += 4) {
      v2f a = afrag(B2, SZ, mb, k, lane);
      v2f b = bfrag(Wc2, 60, k, nb, lane, 60);
      a_ = wmma4(a, b, a_);
    }
    __syncthreads();
    store_tile(B0, SZ2, mb, nb, lane, a_, bc2, 60, NCLS, true);  // z2 -> B0
  }
  __syncthreads();

  // ============ final: out[c] = z2[c,:] . Wc3 + bc3 ========================
  if (tid < NCLS) {
    float s = bc3[0];
    const float* zr = B0 + tid * SZ2;
    for (int k = 0; k < 60; ++k) s += zr[k] * Wc3[k];
    out[(size_t)blk * NCLS + tid] = s;
  }
}

extern "C" void kernel_launch(void* const* d_in, const int* in_sizes, int n_in,
                              void* d_out, int out_size, void* d_ws,
                              size_t ws_size, hipStream_t stream) {
  (void)in_sizes; (void)n_in; (void)out_size; (void)d_ws; (void)ws_size;
  const float* x1  = (const float*)d_in[0];
  const int*   ei1 = (const int*)d_in[1];
  const float* ea1 = (const float*)d_in[2];
  const float* x2  = (const float*)d_in[3];
  const int*   ei2 = (const int*)d_in[4];
  const float* ea2 = (const float*)d_in[5];
  const float* W1  = (const float*)d_in[6];
  const float* b1  = (const float*)d_in[7];
  const float* W4  = (const float*)d_in[8];
  const float* b4  = (const float*)d_in[9];
  const float* Wc1 = (const float*)d_in[10];
  const float* bc1 = (const float*)d_in[11];
  const float* Wc2 = (const float*)d_in[12];
  const float* bc2 = (const float*)d_in[13];
  const float* Wc3 = (const float*)d_in[14];
  const float* bc3 = (const float*)d_in[15];
  float* out = (float*)d_out;

  dim3 grid(128), block(BLK);
  hipLaunchKernelGGL(siamese_cheb_kernel, grid, block, LDS_BYTES, stream,
                     x1, ei1, ea1, x2, ei2, ea2, W1, b1, W4, b4,
                     Wc1, bc1, Wc2, bc2, Wc3, bc3, out);
}